// QuantumRegression_65481071395299
// MI455X (gfx1250) — compile-verified
//
#include <hip/hip_runtime.h>
#include <math.h>

typedef __attribute__((ext_vector_type(2))) float v2f;
typedef __attribute__((ext_vector_type(8))) float v8f;

#define BATCH   64
#define SEQ     512
#define EMB     4
#define FFN     64
#define NPATCH  (BATCH * SEQ)
#define TPB     256

#if __has_builtin(__builtin_amdgcn_wmma_f32_16x16x4_f32)
#define HAVE_WMMA4 1
static __device__ __forceinline__ v8f wmma4(v2f a, v2f b, v8f c) {
  // D = A(16x4,f32) * B(4x16,f32) + C(16x16,f32)
  return __builtin_amdgcn_wmma_f32_16x16x4_f32(false, a, false, b, (short)0, c,
                                               false, false);
}
#endif

// ---------------------------------------------------------------------------
// Kernel 1: 4-qubit state-vector sim per patch (+ positional encoding)
// ---------------------------------------------------------------------------
__global__ __launch_bounds__(TPB) void qfeat_kernel(
    const float* __restrict__ x, const float* __restrict__ rl,
    float* __restrict__ h0) {
  int pid = blockIdx.x * TPB + threadIdx.x;
  if (pid >= NPATCH) return;

  const float* xp = x + pid * 4;
  float cw[4], sw[4];
#pragma unroll
  for (int w = 0; w < 4; ++w) {
    cw[w] = cosf(0.5f * xp[w]);
    sw[w] = sinf(0.5f * xp[w]);
  }

  float re[16], im[16];
  // Product state after the 4 encoder RY gates (wire w -> bit 3-w).
#pragma unroll
  for (int k = 0; k < 16; ++k) {
    float a = 1.f;
#pragma unroll
    for (int w = 0; w < 4; ++w) a *= ((k >> (3 - w)) & 1) ? sw[w] : cw[w];
    re[k] = a;
    im[k] = 0.f;
  }

  auto ry = [&](float th, int b) {
    float c = cosf(0.5f * th), s = sinf(0.5f * th);
#pragma unroll
    for (int g = 0; g < 8; ++g) {
      int k0 = ((g >> b) << (b + 1)) | (g & ((1 << b) - 1));
      int k1 = k0 | (1 << b);
      float r0 = re[k0], r1 = re[k1], i0 = im[k0], i1 = im[k1];
      re[k0] = c * r0 - s * r1;  im[k0] = c * i0 - s * i1;
      re[k1] = s * r0 + c * r1;  im[k1] = s * i0 + c * i1;
    }
  };
  auto rx = [&](float th, int b) {
    float c = cosf(0.5f * th), s = sinf(0.5f * th);
#pragma unroll
    for (int g = 0; g < 8; ++g) {
      int k0 = ((g >> b) << (b + 1)) | (g & ((1 << b) - 1));
      int k1 = k0 | (1 << b);
      float r0 = re[k0], i0 = im[k0], r1 = re[k1], i1 = im[k1];
      re[k0] = c * r0 + s * i1;  im[k0] = c * i0 - s * r1;
      re[k1] = s * i0 + c * r1;  im[k1] = -s * r0 + c * i1;
    }
  };
  auto rz = [&](float th, int b) {
    float c = cosf(0.5f * th), s = sinf(0.5f * th);
#pragma unroll
    for (int k = 0; k < 16; ++k) {
      float sg = ((k >> b) & 1) ? s : -s;
      float r = re[k], i = im[k];
      re[k] = c * r - sg * i;
      im[k] = c * i + sg * r;
    }
  };
  auto cnotg = [&](int bc, int bt) {
#pragma unroll
    for (int k = 0; k < 16; ++k) {
      if (((k >> bc) & 1) && !((k >> bt) & 1)) {
        int k1 = k | (1 << bt);
        float tr = re[k]; re[k] = re[k1]; re[k1] = tr;
        float ti = im[k]; im[k] = im[k1]; im[k1] = ti;
      }
    }
  };

  float r0 = rl[0], r1v = rl[1], r2 = rl[2], r3 = rl[3], r4 = rl[4];
  ry(r0, 3);        // RY wire0
  rx(r1v, 2);       // RX wire1
  cnotg(3, 2);      // CNOT(0,1)
  rz(r2, 1);        // RZ wire2
  ry(r3, 0);        // RY wire3
  cnotg(1, 0);      // CNOT(2,3)
  rx(r4, 3);        // RX wire0
  cnotg(2, 1);      // CNOT(1,2)

  float f[4] = {0.f, 0.f, 0.f, 0.f};
#pragma unroll
  for (int k = 0; k < 16; ++k) {
    float p = re[k] * re[k] + im[k] * im[k];
#pragma unroll
    for (int w = 0; w < 4; ++w)
      f[w] += ((k >> (3 - w)) & 1) ? -p : p;
  }

  int s = pid & (SEQ - 1);
  float fs = (float)s;
  h0[pid * 4 + 0] = f[0] + sinf(fs);
  h0[pid * 4 + 1] = f[1] + cosf(fs);
  h0[pid * 4 + 2] = f[2] + sinf(0.01f * fs);
  h0[pid * 4 + 3] = f[3] + cosf(0.01f * fs);
}

// ---------------------------------------------------------------------------
// Kernel 2: full transformer (2 blocks) + pool + head. One block per batch b.
// ---------------------------------------------------------------------------
__global__ __launch_bounds__(TPB) void xformer_kernel(
    const float* __restrict__ h0,
    const float* __restrict__ Wq, const float* __restrict__ Wk,
    const float* __restrict__ Wv, const float* __restrict__ Wo,
    const float* __restrict__ bo,
    const float* __restrict__ W1, const float* __restrict__ b1,
    const float* __restrict__ W2, const float* __restrict__ b2,
    const float* __restrict__ g1, const float* __restrict__ be1,
    const float* __restrict__ g2, const float* __restrict__ be2,
    const float* __restrict__ Wh, const float* __restrict__ bh,
    float* __restrict__ out) {
  __shared__ float sh_h[SEQ * EMB];   // 8 KB  : hidden state
  __shared__ float sh_s[8192];        // 32 KB : qkv [512][12]  /  FFN scratch
  __shared__ float sh_t[SEQ * EMB];   // 8 KB  : attn out / FFN out

  const int b    = blockIdx.x;
  const int tid  = threadIdx.x;
  const int lane = tid & 31;
  const int wave = tid >> 5;
  const int n    = lane & 15;            // WMMA N column / A row index
  const int kb   = (lane >> 4) << 1;     // WMMA K base (0 or 2)

  for (int idx = tid; idx < SEQ * EMB; idx += TPB)
    sh_h[idx] = h0[b * SEQ * EMB + idx];
  __syncthreads();

  for (int l = 0; l < 2; ++l) {
    const float* Wq_l = Wq + l * 16;  const float* Wk_l = Wk + l * 16;
    const float* Wv_l = Wv + l * 16;  const float* Wo_l = Wo + l * 16;
    const float* bo_l = bo + l * 4;
    const float* W1_l = W1 + l * EMB * FFN;  const float* b1_l = b1 + l * FFN;
    const float* W2_l = W2 + l * FFN * EMB;  const float* b2_l = b2 + l * 4;
    const float* g1_l = g1 + l * 4;  const float* be1_l = be1 + l * 4;
    const float* g2_l = g2 + l * 4;  const float* be2_l = be2 + l * 4;

    // ---- 1) Fused QKV projection: h(512x4) @ [Wq|Wk|Wv|0](4x16) ----------
#ifdef HAVE_WMMA4
    {
      v2f bq;
      {
        float v0, v1;
        if (n < 4)       { v0 = Wq_l[kb * 4 + n];        v1 = Wq_l[(kb + 1) * 4 + n]; }
        else if (n < 8)  { v0 = Wk_l[kb * 4 + (n - 4)];  v1 = Wk_l[(kb + 1) * 4 + (n - 4)]; }
        else if (n < 12) { v0 = Wv_l[kb * 4 + (n - 8)];  v1 = Wv_l[(kb + 1) * 4 + (n - 8)]; }
        else             { v0 = 0.f;                     v1 = 0.f; }
        bq.x = v0; bq.y = v1;
      }
      for (int tile = wave; tile < SEQ / 16; tile += 8) {
        int row = tile * 16 + n;
        v2f a; a.x = sh_h[row * 4 + kb]; a.y = sh_h[row * 4 + kb + 1];
        v8f c = {};
        v8f d = wmma4(a, bq, c);
        int rbase = tile * 16 + ((lane >> 4) << 3);
        if (n < 12) {
#pragma unroll
          for (int g = 0; g < 8; ++g) sh_s[(rbase + g) * 12 + n] = d[g];
        }
      }
    }
#else
    for (int r = tid; r < SEQ; r += TPB) {
      float hv[4];
#pragma unroll
      for (int d = 0; d < 4; ++d) hv[d] = sh_h[r * 4 + d];
#pragma unroll
      for (int c = 0; c < 4; ++c) {
        float q = 0.f, k = 0.f, v = 0.f;
#pragma unroll
        for (int e = 0; e < 4; ++e) {
          q += hv[e] * Wq_l[e * 4 + c];
          k += hv[e] * Wk_l[e * 4 + c];
          v += hv[e] * Wv_l[e * 4 + c];
        }
        sh_s[r * 12 + c] = q; sh_s[r * 12 + 4 + c] = k; sh_s[r * 12 + 8 + c] = v;
      }
    }
#endif
    __syncthreads();

    // ---- 2) Attention (dk=1, scale=1): online softmax over j -------------
    for (int t = tid; t < 4 * SEQ; t += TPB) {
      int head = t >> 9;
      int i    = t & (SEQ - 1);
      float qi = sh_s[i * 12 + head];
      float m = -3.0e38f, lsum = 0.f, acc = 0.f;
#pragma unroll 4
      for (int j = 0; j < SEQ; ++j) {
        float sc = qi * sh_s[j * 12 + 4 + head];
        if (sc > m) {
          float corr = __expf(m - sc);
          lsum *= corr; acc *= corr; m = sc;
        }
        float e = __expf(sc - m);
        lsum += e;
        acc += e * sh_s[j * 12 + 8 + head];
      }
      sh_t[i * 4 + head] = acc / lsum;
    }
    __syncthreads();

    // ---- 3) residual + o@Wo + bo, LayerNorm1 -----------------------------
    for (int r = tid; r < SEQ; r += TPB) {
      float o0 = sh_t[r * 4 + 0], o1 = sh_t[r * 4 + 1];
      float o2 = sh_t[r * 4 + 2], o3 = sh_t[r * 4 + 3];
      float ov[4];
#pragma unroll
      for (int d = 0; d < 4; ++d)
        ov[d] = sh_h[r * 4 + d] + o0 * Wo_l[d] + o1 * Wo_l[4 + d] +
                o2 * Wo_l[8 + d] + o3 * Wo_l[12 + d] + bo_l[d];
      float mean = 0.25f * (ov[0] + ov[1] + ov[2] + ov[3]);
      float var = 0.f;
#pragma unroll
      for (int d = 0; d < 4; ++d) { float q = ov[d] - mean; var += q * q; }
      var *= 0.25f;
      float inv = rsqrtf(var + 1e-5f);
#pragma unroll
      for (int d = 0; d < 4; ++d)
        sh_h[r * 4 + d] = (ov[d] - mean) * inv * g1_l[d] + be1_l[d];
    }
    __syncthreads();

    // ---- 4) FFN: relu(h@W1+b1)@W2 + b2 -----------------------------------
#ifdef HAVE_WMMA4
    {
      float* fb = sh_s + wave * 1024;  // per-wave [16][64] scratch
      for (int tile = wave; tile < SEQ / 16; tile += 8) {
        int row = tile * 16 + n;
        v2f a; a.x = sh_h[row * 4 + kb]; a.y = sh_h[row * 4 + kb + 1];
        int rb = (lane >> 4) << 3;
#pragma unroll
        for (int nt = 0; nt < 4; ++nt) {
          v2f bb;
          bb.x = W1_l[kb * FFN + nt * 16 + n];
          bb.y = W1_l[(kb + 1) * FFN + nt * 16 + n];
          v8f c = {};
          v8f d = wmma4(a, bb, c);
          float bias = b1_l[nt * 16 + n];
#pragma unroll
          for (int g = 0; g < 8; ++g) {
            float v = d[g] + bias;
            fb[(rb + g) * FFN + nt * 16 + n] = v > 0.f ? v : 0.f;
          }
        }
        // second GEMM: f(16x64) @ W2(64x4, N padded to 16), K in 16 chunks
        v8f c2 = {};
#pragma unroll
        for (int ck = 0; ck < 16; ++ck) {
          v2f a2;
          a2.x = fb[n * FFN + ck * 4 + kb];
          a2.y = fb[n * FFN + ck * 4 + kb + 1];
          v2f b2v;
          b2v.x = (n < 4) ? W2_l[(ck * 4 + kb) * 4 + n] : 0.f;
          b2v.y = (n < 4) ? W2_l[(ck * 4 + kb + 1) * 4 + n] : 0.f;
          c2 = wmma4(a2, b2v, c2);
        }
        int obase = tile * 16 + rb;
        if (n < 4) {
          float bias = b2_l[n];
#pragma unroll
          for (int g = 0; g < 8; ++g) sh_t[(obase + g) * 4 + n] = c2[g] + bias;
        }
      }
    }
#else
    for (int r = tid; r < SEQ; r += TPB) {
      float hv[4];
#pragma unroll
      for (int d = 0; d < 4; ++d) hv[d] = sh_h[r * 4 + d];
      float o[4];
#pragma unroll
      for (int d = 0; d < 4; ++d) o[d] = b2_l[d];
      for (int j = 0; j < FFN; ++j) {
        float f = b1_l[j];
#pragma unroll
        for (int e = 0; e < 4; ++e) f += hv[e] * W1_l[e * FFN + j];
        f = f > 0.f ? f : 0.f;
#pragma unroll
        for (int d = 0; d < 4; ++d) o[d] += f * W2_l[j * 4 + d];
      }
#pragma unroll
      for (int d = 0; d < 4; ++d) sh_t[r * 4 + d] = o[d];
    }
#endif
    __syncthreads();

    // ---- 5) residual + LayerNorm2 ----------------------------------------
    for (int r = tid; r < SEQ; r += TPB) {
      float xv[4];
#pragma unroll
      for (int d = 0; d < 4; ++d) xv[d] = sh_h[r * 4 + d] + sh_t[r * 4 + d];
      float mean = 0.25f * (xv[0] + xv[1] + xv[2] + xv[3]);
      float var = 0.f;
#pragma unroll
      for (int d = 0; d < 4; ++d) { float q = xv[d] - mean; var += q * q; }
      var *= 0.25f;
      float inv = rsqrtf(var + 1e-5f);
#pragma unroll
      for (int d = 0; d < 4; ++d)
        sh_h[r * 4 + d] = (xv[d] - mean) * inv * g2_l[d] + be2_l[d];
    }
    __syncthreads();
  }

  // ---- mean pool + head ---------------------------------------------------
  if (tid < 4) sh_s[tid] = 0.f;
  __syncthreads();
  {
    float part[4] = {0.f, 0.f, 0.f, 0.f};
    for (int r = tid; r < SEQ; r += TPB) {
#pragma unroll
      for (int d = 0; d < 4; ++d) part[d] += sh_h[r * 4 + d];
    }
#pragma unroll
    for (int d = 0; d < 4; ++d) atomicAdd(&sh_s[d], part[d]);
  }
  __syncthreads();
  if (tid == 0) {
    float acc = bh[0];
#pragma unroll
    for (int d = 0; d < 4; ++d) acc += (sh_s[d] * (1.f / (float)SEQ)) * Wh[d];
    out[b] = acc;
  }
}

// ---------------------------------------------------------------------------
extern "C" void kernel_launch(void* const* d_in, const int* in_sizes, int n_in,
                              void* d_out, int out_size, void* d_ws,
                              size_t ws_size, hipStream_t stream) {
  const float* x   = (const float*)d_in[0];
  const float* rl  = (const float*)d_in[1];
  const float* Wq  = (const float*)d_in[2];
  const float* Wk  = (const float*)d_in[3];
  const float* Wv  = (const float*)d_in[4];
  const float* Wo  = (const float*)d_in[5];
  const float* bo  = (const float*)d_in[6];
  const float* W1  = (const float*)d_in[7];
  const float* b1  = (const float*)d_in[8];
  const float* W2  = (const float*)d_in[9];
  const float* b2  = (const float*)d_in[10];
  const float* g1  = (const float*)d_in[11];
  const float* be1 = (const float*)d_in[12];
  const float* g2  = (const float*)d_in[13];
  const float* be2 = (const float*)d_in[14];
  const float* Wh  = (const float*)d_in[15];
  const float* bh  = (const float*)d_in[16];
  float* out = (float*)d_out;
  float* h0  = (float*)d_ws;  // 64*512*4 floats = 512 KB

  qfeat_kernel<<<(NPATCH + TPB - 1) / TPB, TPB, 0, stream>>>(x, rl, h0);
  xformer_kernel<<<BATCH, TPB, 0, stream>>>(h0, Wq, Wk, Wv, Wo, bo, W1, b1, W2,
                                            b2, g1, be1, g2, be2, Wh, bh, out);
}